// GNNActor_47605417509063
// MI455X (gfx1250) — compile-verified
//
#include <hip/hip_runtime.h>
#include <math.h>

#define NNODES 300000
#define NEDGES 1200000
#define CCH    128
#define HH     32

typedef float v2f __attribute__((ext_vector_type(2)));
typedef float v8f __attribute__((ext_vector_type(8)));

// workspace layout (in floats)
#define OFF_Y    0ull
#define OFF_AGG  38400000ull   // NNODES*128
#define OFF_DEG  76800000ull   // deg -> dinv in place
#define OFF_CONC 77100000ull
#define OFF_SUM  77400000ull

// ---------------- init: zero agg, deg=1 (self loop), sum=0 ----------------
__global__ void k_init(float* __restrict__ ws) {
  long long gid = (long long)blockIdx.x * 256 + threadIdx.x;
  if (gid < (long long)NNODES * CCH) ws[OFF_AGG + gid] = 0.0f;
  if (gid < NNODES)                  ws[OFF_DEG + gid] = 1.0f;
  if (gid == 0)                      ws[OFF_SUM] = 0.0f;
}

// ---------------- degree: deg[dst] += 1 over edges ----------------
__global__ void k_degree(const long long* __restrict__ ei, float* __restrict__ deg) {
  int e = blockIdx.x * 256 + threadIdx.x;
  if (e < NEDGES) {
    int d = (int)ei[NEDGES + e];   // row 1 = dst
    atomicAdd(&deg[d], 1.0f);
  }
}

// ---------------- dinv = rsqrt(deg), deg >= 1 always ----------------
__global__ void k_rsqrt(float* __restrict__ deg) {
  int i = blockIdx.x * 256 + threadIdx.x;
  if (i < NNODES) deg[i] = rsqrtf(deg[i]);
}

// ---------------- y = dinv[:,None] * (state @ conv_w), fp32 WMMA ----------------
// block: 256 threads (8 waves), computes 128 rows x 128 cols.
__global__ void __launch_bounds__(256)
k_gemm_scale(const float* __restrict__ x, const float* __restrict__ w,
             const float* __restrict__ dinv, float* __restrict__ y) {
  extern __shared__ float lds[];
  float* As = lds;               // [128][132] row-major A tile
  float* Ws = lds + 128 * 132;   // [128][132] = W transposed: Ws[n*132+k] = w[k*128+n]

  const int t = threadIdx.x;
  const int R = blockIdx.x * 128;

  for (int i = 0; i < 64; ++i) {
    int idx = i * 256 + t;              // 0..16383
    int k = idx >> 7, n = idx & 127;
    Ws[n * 132 + k] = w[idx];
  }
  for (int i = 0; i < 16; ++i) {
    int fid = i * 256 + t;              // 0..4095
    int r = fid >> 5, c4 = fid & 31;
    int row = R + r;
    float4 v = make_float4(0.f, 0.f, 0.f, 0.f);
    if (row < NNODES) v = ((const float4*)x)[(long long)row * 32 + c4];
    *(float4*)&As[r * 132 + c4 * 4] = v;
  }
  __syncthreads();

  const int wv   = t >> 5;
  const int lane = t & 31;
  const int ml   = lane & 15;
  const int kh   = lane >> 4;
  const int m    = wv * 16 + ml;

  v2f a[32];
#pragma unroll
  for (int ks = 0; ks < 32; ++ks)
    a[ks] = *(const v2f*)&As[m * 132 + ks * 4 + kh * 2];

  float dv[8];
#pragma unroll
  for (int r = 0; r < 8; ++r) {
    int row = R + wv * 16 + r + kh * 8;
    dv[r] = (row < NNODES) ? dinv[row] : 0.0f;
  }

  for (int nt = 0; nt < 8; ++nt) {
    int n = nt * 16 + ml;
    v8f acc = {0.f, 0.f, 0.f, 0.f, 0.f, 0.f, 0.f, 0.f};
#pragma unroll
    for (int ks = 0; ks < 32; ++ks) {
      v2f b = *(const v2f*)&Ws[n * 132 + ks * 4 + kh * 2];
      acc = __builtin_amdgcn_wmma_f32_16x16x4_f32(
          false, a[ks], false, b, (short)0, acc, false, false);
    }
#pragma unroll
    for (int r = 0; r < 8; ++r) {
      int row = R + wv * 16 + r + kh * 8;
      if (row < NNODES) y[(long long)row * CCH + n] = dv[r] * acc[r];
    }
  }
}

// ---------------- scatter: agg[dst] += y[src], one wave per edge ----------------
__global__ void k_scatter(const long long* __restrict__ ei,
                          const float* __restrict__ y, float* __restrict__ agg) {
  long long gid = (long long)blockIdx.x * 256 + threadIdx.x;
  int e    = (int)(gid >> 5);
  int lane = (int)(gid & 31);
  if (e < NEDGES) {
    int s = (int)ei[e];
    int d = (int)ei[NEDGES + e];
    const float* ys = y + (long long)s * CCH;
    float* ad = agg + (long long)d * CCH;
#pragma unroll
    for (int q = 0; q < 4; ++q) {
      int c = lane + q * 32;
      atomicAdd(&ad[c], ys[c]);
    }
  }
}

// ---------------- fused GCN epilogue + WMMA MLP head ----------------
// block: 256 threads (8 waves) handles 128 nodes.
__global__ void __launch_bounds__(256)
k_mlp(const float* __restrict__ x, const float* __restrict__ y,
      const float* __restrict__ agg, const float* __restrict__ dinv,
      const float* __restrict__ convb,
      const float* __restrict__ w1, const float* __restrict__ b1,
      const float* __restrict__ w2, const float* __restrict__ b2,
      const float* __restrict__ w3, const float* __restrict__ b3,
      float* __restrict__ conc, float* __restrict__ gsum) {
  extern __shared__ float lds[];
  float* hs  = lds;              // [128][132]  h matrix
  float* w1t = hs  + 16896;      // [32][132]   w1 transposed
  float* x1s = w1t + 4224;       // [128][36]
  float* x2s = x1s + 4608;       // [128][36]
  float* w2t = x2s + 4608;       // [32][36]    w2 transposed
  float* b1s = w2t + 1152;       // [32]
  float* b2s = b1s + 32;         // [32]
  float* w3s = b2s + 32;         // [32]
  float* b3s = w3s + 32;         // [1]

  const int t  = threadIdx.x;
  const int B0 = blockIdx.x * 128;

  // stage h = relu(dinv*(agg+y)+conv_b) + state, zero-pad tail nodes
  for (int i = 0; i < 64; ++i) {
    int idx = i * 256 + t;       // 0..16383
    int r = idx >> 7, c = idx & 127;
    int node = B0 + r;
    float v = 0.0f;
    if (node < NNODES) {
      long long g = (long long)node * CCH + c;
      float o = dinv[node] * (agg[g] + y[g]) + convb[c];
      o = o > 0.f ? o : 0.f;
      v = o + x[g];
    }
    hs[r * 132 + c] = v;
  }
  // stage w1^T, w2^T, biases
  for (int i = 0; i < 16; ++i) {
    int idx = i * 256 + t;       // 0..4095
    int k = idx >> 5, j = idx & 31;
    w1t[j * 132 + k] = w1[idx];
  }
  for (int i = 0; i < 4; ++i) {
    int idx = i * 256 + t;       // 0..1023
    int k = idx >> 5, j = idx & 31;
    w2t[j * 36 + k] = w2[idx];
  }
  if (t < HH) { b1s[t] = b1[t]; b2s[t] = b2[t]; w3s[t] = w3[t]; }
  if (t == 0) b3s[0] = b3[0];
  __syncthreads();

  const int wv   = t >> 5;
  const int lane = t & 31;
  const int ml   = lane & 15;
  const int kh   = lane >> 4;
  const int m    = wv * 16 + ml;

  // ---- layer 1: x1 = leaky_relu(h @ w1 + b1), M=128 K=128 N=32 ----
  {
    v2f a[32];
#pragma unroll
    for (int ks = 0; ks < 32; ++ks)
      a[ks] = *(const v2f*)&hs[m * 132 + ks * 4 + kh * 2];
    for (int nt = 0; nt < 2; ++nt) {
      int n = nt * 16 + ml;
      float bv = b1s[n];
      v8f acc = {bv, bv, bv, bv, bv, bv, bv, bv};
#pragma unroll
      for (int ks = 0; ks < 32; ++ks) {
        v2f b = *(const v2f*)&w1t[n * 132 + ks * 4 + kh * 2];
        acc = __builtin_amdgcn_wmma_f32_16x16x4_f32(
            false, a[ks], false, b, (short)0, acc, false, false);
      }
#pragma unroll
      for (int r = 0; r < 8; ++r) {
        int row = wv * 16 + r + kh * 8;
        float v = acc[r];
        v = v > 0.f ? v : 0.01f * v;
        x1s[row * 36 + n] = v;
      }
    }
  }
  __syncthreads();

  // ---- layer 2: x2 = leaky_relu(x1 @ w2 + b2), M=128 K=32 N=32 ----
  {
    v2f a[8];
#pragma unroll
    for (int ks = 0; ks < 8; ++ks)
      a[ks] = *(const v2f*)&x1s[m * 36 + ks * 4 + kh * 2];
    for (int nt = 0; nt < 2; ++nt) {
      int n = nt * 16 + ml;
      float bv = b2s[n];
      v8f acc = {bv, bv, bv, bv, bv, bv, bv, bv};
#pragma unroll
      for (int ks = 0; ks < 8; ++ks) {
        v2f b = *(const v2f*)&w2t[n * 36 + ks * 4 + kh * 2];
        acc = __builtin_amdgcn_wmma_f32_16x16x4_f32(
            false, a[ks], false, b, (short)0, acc, false, false);
      }
#pragma unroll
      for (int r = 0; r < 8; ++r) {
        int row = wv * 16 + r + kh * 8;
        float v = acc[r];
        v = v > 0.f ? v : 0.01f * v;
        x2s[row * 36 + n] = v;
      }
    }
  }
  __syncthreads();

  // ---- layer 3: softplus(x2 @ w3 + b3), one thread per node ----
  if (t < 128) {
    int node = B0 + t;
    if (node < NNODES) {
      float s = b3s[0];
#pragma unroll
      for (int j = 0; j < HH; ++j) s += x2s[t * 36 + j] * w3s[j];
      float c = (s > 20.f) ? s : log1pf(expf(s));
      conc[node] = c;
      atomicAdd(gsum, c);
    }
  }
}

// ---------------- normalize ----------------
__global__ void k_final(const float* __restrict__ conc, const float* __restrict__ gsum,
                        float* __restrict__ out) {
  int i = blockIdx.x * 256 + threadIdx.x;
  if (i < NNODES) out[i] = conc[i] / (gsum[0] + 1e-20f);
}

extern "C" void kernel_launch(void* const* d_in, const int* in_sizes, int n_in,
                              void* d_out, int out_size, void* d_ws, size_t ws_size,
                              hipStream_t stream) {
  const float*     state  = (const float*)d_in[0];
  const long long* ei     = (const long long*)d_in[1];
  const float*     conv_w = (const float*)d_in[2];
  const float*     conv_b = (const float*)d_in[3];
  const float*     w1     = (const float*)d_in[4];
  const float*     b1     = (const float*)d_in[5];
  const float*     w2     = (const float*)d_in[6];
  const float*     b2     = (const float*)d_in[7];
  const float*     w3     = (const float*)d_in[8];
  const float*     b3     = (const float*)d_in[9];

  float* ws   = (float*)d_ws;
  float* y    = ws + OFF_Y;
  float* agg  = ws + OFF_AGG;
  float* deg  = ws + OFF_DEG;   // becomes dinv in place
  float* conc = ws + OFF_CONC;
  float* gsum = ws + OFF_SUM;
  float* out  = (float*)d_out;

  k_init  <<<150000, 256, 0, stream>>>(ws);
  k_degree<<<(NEDGES + 255) / 256, 256, 0, stream>>>(ei, deg);
  k_rsqrt <<<(NNODES + 255) / 256, 256, 0, stream>>>(deg);

  size_t shmem_g = (size_t)2 * 128 * 132 * sizeof(float);   // A tile + W^T
  k_gemm_scale<<<(NNODES + 127) / 128, 256, shmem_g, stream>>>(state, conv_w, deg, y);

  k_scatter<<<(int)(((long long)NEDGES * 32 + 255) / 256), 256, 0, stream>>>(ei, y, agg);

  size_t shmem_m = (size_t)(16896 + 4224 + 4608 + 4608 + 1152 + 97) * sizeof(float);
  k_mlp<<<(NNODES + 127) / 128, 256, shmem_m, stream>>>(state, y, agg, deg, conv_b,
                                                        w1, b1, w2, b2, w3, b3, conc, gsum);
  k_final<<<(NNODES + 255) / 256, 256, 0, stream>>>(conc, gsum, out);
}